// NGCF_1176821039769
// MI455X (gfx1250) — compile-verified
//
#include <hip/hip_runtime.h>

#define N_USERS 60000
#define N_ITEMS 40000
#define NTOT    100000
#define EMB     64
#define NNZ     2560000
#define BATCH   4096
#define NLAYERS 3

typedef __attribute__((ext_vector_type(2))) float v2f;
typedef __attribute__((ext_vector_type(8))) float v8f;

// ---------------------------------------------------------------------------
// 1) Build masked ego embeddings into columns [0:64) of the concat buffer.
// ---------------------------------------------------------------------------
__global__ void ngcf_init_mask(const float* __restrict__ uemb,
                               const float* __restrict__ iemb,
                               const int* __restrict__ usz,
                               const int* __restrict__ isz,
                               float* __restrict__ allemb) {
    int idx = blockIdx.x * blockDim.x + threadIdx.x;
    if (idx >= NTOT * EMB) return;
    int i = idx >> 6, d = idx & 63;
    float v; int sz;
    if (i < N_USERS) { v = uemb[(size_t)i * EMB + d]; sz = usz[i]; }
    else             { int j = i - N_USERS; v = iemb[(size_t)j * EMB + d]; sz = isz[j]; }
    allemb[(size_t)i * 256 + d] = (d < sz) ? v : 0.0f;
}

// ---------------------------------------------------------------------------
// 2) COO SpMM: side[row] += val * ego[col].  16 lanes per edge, float4 gather,
//    f32 atomic scatter (working set is L2-resident: ego/side = 25.6MB each).
// ---------------------------------------------------------------------------
__global__ void ngcf_spmm(const float* __restrict__ vals,
                          const int* __restrict__ rows,
                          const int* __restrict__ cols,
                          const float* __restrict__ ego, int ldE,
                          float* __restrict__ side) {
    int t = blockIdx.x * blockDim.x + threadIdx.x;
    int e = t >> 4;
    if (e >= NNZ) return;
    int lane = t & 15;
    float val = vals[e];
    int r = rows[e], c = cols[e];
    float4 v = ((const float4*)(ego + (size_t)c * ldE))[lane];
    float* dst = side + (size_t)r * EMB + lane * 4;
    atomicAdd(dst + 0, val * v.x);
    atomicAdd(dst + 1, val * v.y);
    atomicAdd(dst + 2, val * v.z);
    atomicAdd(dst + 3, val * v.w);
}

// ---------------------------------------------------------------------------
// 3) Per-layer fused kernel: one wave owns a 16-row tile.
//    sum = side @ Wgc ; bi = (ego*side) @ Wbi   via V_WMMA_F32_16X16X4_F32
//    act = leaky_relu(sum + bi + bgc + bbi)  -> actout (feeds next layer)
//    act / max(||act||_2, 1e-12)             -> normout (concat buffer)
// ---------------------------------------------------------------------------
__global__ void __launch_bounds__(256)
ngcf_layer(const float* __restrict__ side, const float* __restrict__ ego, int ldE,
           const float* __restrict__ Wgc, const float* __restrict__ bgc,
           const float* __restrict__ Wbi, const float* __restrict__ bbi,
           float* __restrict__ actout, float* __restrict__ normout) {
    const int NTILES = NTOT / 16;                       // 6250
    int wave = (blockIdx.x * blockDim.x + threadIdx.x) >> 5;
    if (wave >= NTILES) return;                         // wave-uniform guard (EXEC all-1 inside)
    int lane = threadIdx.x & 31;
    int lo = lane & 15;
    int hi = lane >> 4;
    int row0 = wave * 16;

    // A-fragment source rows: lane lo owns matrix row (row0+lo); hi selects K sub-pair
    const float* arow_s = side + (size_t)(row0 + lo) * EMB + 2 * hi;
    const float* arow_e = ego  + (size_t)(row0 + lo) * ldE + 2 * hi;

    v8f accS[4] = {{}, {}, {}, {}};
    v8f accB[4] = {{}, {}, {}, {}};

    for (int kk = 0; kk < 16; ++kk) {                   // K = 64 in chunks of 4
        float s0 = arow_s[4 * kk], s1 = arow_s[4 * kk + 1];
        float e0 = arow_e[4 * kk], e1 = arow_e[4 * kk + 1];
        v2f aS; aS.x = s0;      aS.y = s1;              // side fragment
        v2f aP; aP.x = s0 * e0; aP.y = s1 * e1;         // (ego*side) fragment
        const float* wg = Wgc + (size_t)(4 * kk + 2 * hi) * EMB + lo;
        const float* wb = Wbi + (size_t)(4 * kk + 2 * hi) * EMB + lo;
#pragma unroll
        for (int nt = 0; nt < 4; ++nt) {                // 4 column tiles of 16
            v2f bG; bG.x = wg[nt * 16]; bG.y = wg[EMB + nt * 16];
            v2f bB; bB.x = wb[nt * 16]; bB.y = wb[EMB + nt * 16];
            accS[nt] = __builtin_amdgcn_wmma_f32_16x16x4_f32(
                false, aS, false, bG, (short)0, accS[nt], false, false);
            accB[nt] = __builtin_amdgcn_wmma_f32_16x16x4_f32(
                false, aP, false, bB, (short)0, accB[nt], false, false);
        }
    }

    // bias + leaky_relu(0.2) + per-row sum of squares
    float act[4][8];
    float ss[8];
#pragma unroll
    for (int v = 0; v < 8; ++v) ss[v] = 0.0f;
#pragma unroll
    for (int nt = 0; nt < 4; ++nt) {
        float bias = bgc[nt * 16 + lo] + bbi[nt * 16 + lo];
#pragma unroll
        for (int v = 0; v < 8; ++v) {
            float x = accS[nt][v] + accB[nt][v] + bias;
            x = (x >= 0.0f) ? x : 0.2f * x;
            act[nt][v] = x;
            ss[v] += x * x;
        }
    }
    // reduce across the 16 lanes that share each row (xor<16 stays in half-wave)
#pragma unroll
    for (int v = 0; v < 8; ++v) {
        float s = ss[v];
        s += __shfl_xor(s, 1, 32);
        s += __shfl_xor(s, 2, 32);
        s += __shfl_xor(s, 4, 32);
        s += __shfl_xor(s, 8, 32);
        ss[v] = 1.0f / fmaxf(sqrtf(s), 1e-12f);
    }
    // C layout: VGPR v, lanes0-15 -> row v ; lanes16-31 -> row v+8 ; col = 16*nt+lo
#pragma unroll
    for (int nt = 0; nt < 4; ++nt) {
#pragma unroll
        for (int v = 0; v < 8; ++v) {
            int grow = row0 + v + 8 * hi;
            int col  = nt * 16 + lo;
            actout [(size_t)grow * EMB + col] = act[nt][v];
            normout[(size_t)grow * 256 + col] = act[nt][v] * ss[v];
        }
    }
}

// ---------------------------------------------------------------------------
// 4) Final gather of (users, pos, neg) rows from the [N,256] concat buffer.
// ---------------------------------------------------------------------------
__global__ void ngcf_gather(const float* __restrict__ allemb,
                            const int* __restrict__ users,
                            const int* __restrict__ pos,
                            const int* __restrict__ neg,
                            float* __restrict__ out) {
    int t = blockIdx.x * blockDim.x + threadIdx.x;
    if (t >= 3 * BATCH * 64) return;
    int r = t >> 6, q = t & 63;
    int grp = r / BATCH, b = r - grp * BATCH;
    int src = (grp == 0) ? users[b]
            : (grp == 1) ? (N_USERS + pos[b])
                         : (N_USERS + neg[b]);
    float4 v = ((const float4*)(allemb + (size_t)src * 256))[q];
    ((float4*)(out + (size_t)r * 256))[q] = v;
}

// ---------------------------------------------------------------------------
extern "C" void kernel_launch(void* const* d_in, const int* in_sizes, int n_in,
                              void* d_out, int out_size, void* d_ws, size_t ws_size,
                              hipStream_t stream) {
    (void)in_sizes; (void)n_in; (void)out_size; (void)ws_size;
    const float* uemb = (const float*)d_in[0];
    const float* iemb = (const float*)d_in[1];
    const float* Wgc  = (const float*)d_in[2];
    const float* bgc  = (const float*)d_in[3];
    const float* Wbi  = (const float*)d_in[4];
    const float* bbi  = (const float*)d_in[5];
    const float* vals = (const float*)d_in[6];
    const int*   rows = (const int*)d_in[7];
    const int*   cols = (const int*)d_in[8];
    const int*   usz  = (const int*)d_in[9];
    const int*   isz  = (const int*)d_in[10];
    const int*   users = (const int*)d_in[11];
    const int*   pos   = (const int*)d_in[12];
    const int*   neg   = (const int*)d_in[13];
    float* out = (float*)d_out;

    // workspace layout (floats): concat[N,256] | side[N,64] | actA[N,64] | actB[N,64]
    float* ws     = (float*)d_ws;
    float* allemb = ws;
    float* side   = allemb + (size_t)NTOT * 256;
    float* actA   = side   + (size_t)NTOT * EMB;
    float* actB   = actA   + (size_t)NTOT * EMB;

    ngcf_init_mask<<<(NTOT * EMB + 255) / 256, 256, 0, stream>>>(uemb, iemb, usz, isz, allemb);

    const float* ego = allemb;
    int ldE = 256;
    float* actbufs[2] = {actA, actB};
    for (int k = 0; k < NLAYERS; ++k) {
        hipMemsetAsync(side, 0, (size_t)NTOT * EMB * sizeof(float), stream);
        int spmm_threads = NNZ * 16;
        ngcf_spmm<<<(spmm_threads + 255) / 256, 256, 0, stream>>>(vals, rows, cols, ego, ldE, side);
        float* actout = actbufs[k & 1];
        int nblk = ((NTOT / 16) * 32 + 255) / 256;
        ngcf_layer<<<nblk, 256, 0, stream>>>(side, ego, ldE,
                                             Wgc + (size_t)k * EMB * EMB,
                                             bgc + (size_t)k * EMB,
                                             Wbi + (size_t)k * EMB * EMB,
                                             bbi + (size_t)k * EMB,
                                             actout, allemb + (size_t)(k + 1) * EMB);
        ego = actout;
        ldE = EMB;
    }
    ngcf_gather<<<(3 * BATCH * 64 + 255) / 256, 256, 0, stream>>>(allemb, users, pos, neg, out);
}